// VQBlock_21174188769555
// MI455X (gfx1250) — compile-verified
//
#include <hip/hip_runtime.h>

typedef __attribute__((ext_vector_type(2))) float v2f;
typedef __attribute__((ext_vector_type(8))) float v8f;

#define NUM_CODES 512
#define DIM 64
#define LDS_STRIDE 68   // 64 data + 1 bias (-0.5||d||^2) + 3 pad -> conflict-free b64 frag loads
#define N_VECS (16*128*128)

__global__ __launch_bounds__(256)
void vq_main(const float* __restrict__ x, const float* __restrict__ dict,
             float* __restrict__ out, float* __restrict__ partial) {
  extern __shared__ float smem[];                 // dictT[512][68] + 8 wave sums
  float* dictT = smem;
  float* wsum  = smem + NUM_CODES*LDS_STRIDE;

  const int tid  = threadIdx.x;
  const int lane = tid & 31;
  const int wave = tid >> 5;
  const int lm   = lane & 15;   // row / code-in-tile selector
  const int h    = lane >> 4;   // lane half -> K-pair selector

  // ---- Stage 1: dictionary -> LDS transposed, append -0.5*||d||^2 bias column ----
  for (int code = tid; code < NUM_CODES; code += 256) {
    float nrm = 0.f;
    float* row = &dictT[code*LDS_STRIDE];
#pragma unroll 8
    for (int d = 0; d < DIM; ++d) {
      float v = dict[d*NUM_CODES + code];   // coalesced across lanes (consecutive codes)
      row[d] = v;
      nrm += v*v;
    }
    row[64] = -0.5f*nrm; row[65] = 0.f; row[66] = 0.f; row[67] = 0.f;
  }
  __syncthreads();

  // ---- Stage 2: this lane's x fragments (f32 16x4 operand layout), loaded ONCE ----
  const int myrow = blockIdx.x*128 + wave*16 + lm;
  const float* xrow = x + (size_t)myrow*DIM;
  v2f xf[17];
#pragma unroll
  for (int s = 0; s < 16; ++s)
    xf[s] = *(const v2f*)(xrow + 4*s + 2*h);
  // bias K-step: B[64][*]=1 picks up the -0.5||d||^2 column of A
  v2f xb; xb.x = (h == 0) ? 1.f : 0.f; xb.y = 0.f;
  xf[16] = xb;

  // ---- Stage 3: score = f.d - 0.5||d||^2 via WMMA; two interleaved accumulator
  //      chains (tiles c and c+1) to hide dependent-WMMA latency ----
  float maxv[8]; int maxi[8];
#pragma unroll
  for (int r = 0; r < 8; ++r) { maxv[r] = -3.4e38f; maxi[r] = 0; }

  for (int c = 0; c < NUM_CODES/16; c += 2) {
    const float* arow0 = &dictT[((c  )*16 + lm)*LDS_STRIDE + 2*h];
    const float* arow1 = &dictT[((c+1)*16 + lm)*LDS_STRIDE + 2*h];
    v8f acc0 = {0.f,0.f,0.f,0.f,0.f,0.f,0.f,0.f};
    v8f acc1 = {0.f,0.f,0.f,0.f,0.f,0.f,0.f,0.f};
#pragma unroll
    for (int s = 0; s < 17; ++s) {
      v2f a0 = *(const v2f*)(arow0 + 4*s);   // ds_load_2addr_b64-friendly
      v2f a1 = *(const v2f*)(arow1 + 4*s);
      acc0 = __builtin_amdgcn_wmma_f32_16x16x4_f32(false, a0, false, xf[s],
                                                   (short)0, acc0, false, false);
      acc1 = __builtin_amdgcn_wmma_f32_16x16x4_f32(false, a1, false, xf[s],
                                                   (short)0, acc1, false, false);
    }
#pragma unroll
    for (int r = 0; r < 8; ++r) {            // C: vgpr r, lane -> code = 16c + r + 8h
      int code0 = (c  )*16 + r + 8*h;
      int code1 = (c+1)*16 + r + 8*h;
      float v0 = acc0[r];
      bool g0 = v0 > maxv[r];
      maxv[r] = g0 ? v0 : maxv[r];
      maxi[r] = g0 ? code0 : maxi[r];
      float v1 = acc1[r];
      bool g1 = v1 > maxv[r];
      maxv[r] = g1 ? v1 : maxv[r];
      maxi[r] = g1 ? code1 : maxi[r];
    }
  }

  // in-lane reduce over the 8 accumulator slots (ties -> smallest code)
  float bv = maxv[0]; int bi = maxi[0];
#pragma unroll
  for (int r = 1; r < 8; ++r) {
    bool take = (maxv[r] > bv) || ((maxv[r] == bv) && (maxi[r] < bi));
    bv = take ? maxv[r] : bv;
    bi = take ? maxi[r] : bi;
  }
  // combine lane halves: lanes l and l^16 cover the same row lm
  {
    float ov = __shfl_xor(bv, 16, 32);
    int   oi = __shfl_xor(bi, 16, 32);
    bool take = (ov > bv) || ((ov == bv) && (oi < bi));
    bv = take ? ov : bv;
    bi = take ? oi : bi;
  }
  // now every lane holds the winning code for row lm

  // ---- Stage 4: gather q from LDS, write q_st (==q), accumulate loss ----
  const float* qrow = &dictT[bi*LDS_STRIDE];
  float* orow = out + (size_t)myrow*DIM;
  float lossacc = 0.f;
#pragma unroll
  for (int s = 0; s < 16; ++s) {
    v2f q  = *(const v2f*)(qrow + 4*s + 2*h);
    v2f xv = xf[s];
    float d0 = xv.x - q.x, d1 = xv.y - q.y;
    lossacc += d0*d0 + d1*d1;
    *(v2f*)(orow + 4*s + 2*h) = q;
  }
  // deterministic wave -> block loss reduction
#pragma unroll
  for (int m = 16; m >= 1; m >>= 1) lossacc += __shfl_xor(lossacc, m, 32);
  if (lane == 0) wsum[wave] = lossacc;
  __syncthreads();
  if (tid == 0) {
    float t = 0.f;
#pragma unroll
    for (int w = 0; w < 8; ++w) t += wsum[w];
    partial[blockIdx.x] = t;
  }
}

__global__ __launch_bounds__(256)
void vq_reduce(const float* __restrict__ partial, float* __restrict__ lossout, int n) {
  __shared__ float s[256];
  int tid = threadIdx.x;
  float a = 0.f;
  for (int i = tid; i < n; i += 256) a += partial[i];  // fixed order -> deterministic
  s[tid] = a;
  __syncthreads();
  for (int off = 128; off > 0; off >>= 1) {
    if (tid < off) s[tid] += s[tid + off];
    __syncthreads();
  }
  if (tid == 0) lossout[0] = s[0] * 1.25f / (float)(N_VECS * DIM);
}

extern "C" void kernel_launch(void* const* d_in, const int* in_sizes, int n_in,
                              void* d_out, int out_size, void* d_ws, size_t ws_size,
                              hipStream_t stream) {
  const float* xp    = (const float*)d_in[0];
  const float* dictp = (const float*)d_in[1];
  float* out  = (float*)d_out;
  float* part = (float*)d_ws;                       // 2048 floats of scratch

  const int nblocks = N_VECS / 128;                 // 2048 blocks, 8 waves x 16 rows each
  size_t shmem = (size_t)(NUM_CODES*LDS_STRIDE + 8) * sizeof(float);  // ~139.3 KB
  hipFuncSetAttribute(reinterpret_cast<const void*>(vq_main),
                      hipFuncAttributeMaxDynamicSharedMemorySize, (int)shmem);

  hipLaunchKernelGGL(vq_main, dim3(nblocks), dim3(256), shmem, stream,
                     xp, dictp, out, part);
  hipLaunchKernelGGL(vq_reduce, dim3(1), dim3(256), 0, stream,
                     part, out + (size_t)N_VECS*DIM, nblocks);
}